// ChainLoss_28535762715265
// MI455X (gfx1250) — compile-verified
//
#include <hip/hip_runtime.h>

// Problem dims (fixed by the reference setup_inputs()).
constexpr int T = 150;
constexpr int B = 64;
constexpr int D = 2048;
constexpr int S = 20000;
constexpr int A = 262144;

constexpr float NEGV = -1e30f;
constexpr int ARCS_PER_BLOCK = 2048;   // A % ARCS_PER_BLOCK == 0 -> 128 chunks
constexpr int HALF_B = 32;             // one wave = 32 batch lanes (wave32)

// ---------------------------------------------------------------------------
// CDNA5 async global->LDS path (GLOBAL_LOAD_ASYNC_TO_LDS_B128, ASYNCcnt).
// ---------------------------------------------------------------------------
#if defined(__has_builtin)
#if __has_builtin(__builtin_amdgcn_global_load_async_to_lds_b128)
#define HAVE_ASYNC_B128 1
#endif
#if __has_builtin(__builtin_amdgcn_s_wait_asynccnt)
#define HAVE_WAIT_ASYNC 1
#endif
#if __has_builtin(__builtin_amdgcn_wmma_f32_16x16x4_f32)
#define HAVE_WMMA_F32X4 1
#endif
#endif

// Builtin param 1 is 'int __vector_size__(16) __device__ *' per hipcc, i.e. a
// non-const int4* in AS1 (HIP prints AS1 as __device__). Match it exactly.
typedef int v4i_t __attribute__((vector_size(16)));
typedef __attribute__((address_space(1))) v4i_t g_v4i;
typedef __attribute__((address_space(3))) v4i_t l_v4i;

typedef __attribute__((ext_vector_type(2))) float v2f;
typedef __attribute__((ext_vector_type(8))) float v8f;

__device__ __forceinline__ void async_copy16(const float* __restrict__ g, float* l) {
#ifdef HAVE_ASYNC_B128
    // Generic->AS pointers via integer casts: AS1 value == generic value;
    // AS3 value == low 32 bits of the generic LDS address (aperture form).
    __builtin_amdgcn_global_load_async_to_lds_b128(
        (g_v4i*)(unsigned long long)g,
        (l_v4i*)(unsigned int)(unsigned long long)l,
        0, 0);
#else
    *(float4*)l = *(const float4*)g;
#endif
}

__device__ __forceinline__ void wait_async_zero() {
#ifdef HAVE_WAIT_ASYNC
    __builtin_amdgcn_s_wait_asynccnt(0);
#else
#if defined(__AMDGCN__)
    asm volatile("s_wait_asynccnt 0" ::: "memory");
#endif
#endif
}

// Monotonic order-preserving float<->uint encoding for atomic max in log space.
__device__ __forceinline__ unsigned int enc_f32(float f) {
    unsigned int u = __float_as_uint(f);
    return (u & 0x80000000u) ? ~u : (u | 0x80000000u);
}
__device__ __forceinline__ float dec_f32(unsigned int e) {
    unsigned int u = (e & 0x80000000u) ? (e & 0x7FFFFFFFu) : ~e;
    return __uint_as_float(u);
}

// ---------------------------------------------------------------------------
// K0: transpose input [T,B,D] -> inputT [T,D,B]  (so logp gathers are
//     contiguous across batch lanes).
// ---------------------------------------------------------------------------
__global__ __launch_bounds__(256)
void transpose_kernel(const float* __restrict__ in, float* __restrict__ out) {
    __shared__ float tile[32][33];
    const int t  = blockIdx.z;
    const int d0 = blockIdx.x * 32;
    const int b0 = blockIdx.y * 32;
    const float* ip = in  + (size_t)t * B * D;
    float*       op = out + (size_t)t * D * B;
    for (int i = threadIdx.y; i < 32; i += 8)
        tile[i][threadIdx.x] = ip[(size_t)(b0 + i) * D + (d0 + threadIdx.x)];
    __syncthreads();
    for (int i = threadIdx.y; i < 32; i += 8)
        op[(size_t)(d0 + i) * B + (b0 + threadIdx.x)] = tile[threadIdx.x][i];
}

// ---------------------------------------------------------------------------
// K1: alpha[s][b] = init_logp[s];  m = enc(-max);  sum = 0.
// ---------------------------------------------------------------------------
__global__ __launch_bounds__(256)
void init_state_kernel(const float* __restrict__ init_logp,
                       float* __restrict__ alpha,
                       unsigned int* __restrict__ m,
                       float* __restrict__ sum) {
    int i = blockIdx.x * blockDim.x + threadIdx.x;
    if (i < S * B) {
        alpha[i] = init_logp[i >> 6];  // i / B
        m[i]     = 0u;                 // smallest encoded value
        sum[i]   = 0.f;
    }
}

// ---------------------------------------------------------------------------
// K2: arc kernel.  block = (32 lanes = batch half, 8 waves).  Each block:
//   1) async-stage logp_t tile [D][32] (256 KB) into LDS,
//   2) loop over its arc chunk; wave-uniform arc metadata (scalar loads),
//      contiguous 128B gathers/atomics across lanes.
// PASS 0: atomic max of encoded score into m.
// PASS 1: atomic add of exp(score - m) into sum.
// ---------------------------------------------------------------------------
template <int PASS>
__global__ __launch_bounds__(256)
void arc_kernel(const float* __restrict__ logpT_t,      // [D][B] for frame t
                const float* __restrict__ alpha,        // [S][B]
                const float* __restrict__ arc_logw,     // [A]
                const int*   __restrict__ from_state,   // [A]
                const int*   __restrict__ to_state,     // [A]
                const int*   __restrict__ pdf_id,       // [A]
                unsigned int* __restrict__ m,           // [S][B] encoded
                float* __restrict__ sum)                // [S][B]
{
    extern __shared__ float lds_tile[];                 // D * 32 floats = 256 KB
    const int half = blockIdx.y;                        // batch half: 0 or 1
    const int lane = threadIdx.x;                       // 0..31
    const int b    = half * HALF_B + lane;

    // ---- stage logp tile: rows d=0..D-1, 32 floats (128 B) per row ----
    {
        const int tid  = threadIdx.y * 32 + threadIdx.x;  // 0..255
        const int rowi = tid >> 3;                        // 32 rows / iter
        const int quad = (tid & 7) * 4;                   // 16B chunk in row
        for (int it = 0; it < D / 32; ++it) {
            const int d = it * 32 + rowi;
            async_copy16(logpT_t + (size_t)d * B + half * HALF_B + quad,
                         lds_tile + d * HALF_B + quad);
        }
    }
    wait_async_zero();
    __syncthreads();

    // ---- arc loop: wave-uniform arc index, lane = batch ----
    const int base = blockIdx.x * ARCS_PER_BLOCK;
    for (int i = threadIdx.y; i < ARCS_PER_BLOCK; i += 8) {
        const int a   = base + i;
        const int fs  = from_state[a];
        const int ts  = to_state[a];
        const int pdf = pdf_id[a];
        const float w = arc_logw[a];

        const float sc = alpha[(size_t)fs * B + b] + w
                       + lds_tile[pdf * HALF_B + lane];
        const size_t o = (size_t)ts * B + b;
        if (PASS == 0) {
            atomicMax(&m[o], enc_f32(sc));
        } else {
            const float mx = dec_f32(m[o]);              // plain read: pass 0 done
            atomicAdd(&sum[o], __expf(sc - mx));
        }
    }
}

// ---------------------------------------------------------------------------
// K3: alpha = max(m, NEG) + log(sum + 1e-30); reset m/sum for next frame.
// ---------------------------------------------------------------------------
__global__ __launch_bounds__(256)
void finalize_kernel(unsigned int* __restrict__ m,
                     float* __restrict__ sum,
                     float* __restrict__ alpha) {
    int i = blockIdx.x * blockDim.x + threadIdx.x;
    if (i < S * B) {
        const float mx = dec_f32(m[i]);                  // NaN if no arcs hit
        const float s  = sum[i];
        alpha[i] = fmaxf(mx, NEGV) + __logf(s + 1e-30f); // fmaxf drops the NaN
        m[i]   = 0u;
        sum[i] = 0.f;
    }
}

// ---------------------------------------------------------------------------
// K4: per-sequence logsumexp over states.
// ---------------------------------------------------------------------------
__global__ __launch_bounds__(256)
void batch_lse_kernel(const float* __restrict__ alpha,
                      const float* __restrict__ final_logp,
                      float* __restrict__ per_seq) {
    const int b = blockIdx.x;
    const int tid = threadIdx.x;
    __shared__ float red[256];

    float mx = NEGV;
    for (int s = tid; s < S; s += 256)
        mx = fmaxf(mx, alpha[(size_t)s * B + b] + final_logp[s]);
    red[tid] = mx; __syncthreads();
    for (int w = 128; w > 0; w >>= 1) {
        if (tid < w) red[tid] = fmaxf(red[tid], red[tid + w]);
        __syncthreads();
    }
    mx = red[0]; __syncthreads();

    float acc = 0.f;
    for (int s = tid; s < S; s += 256)
        acc += __expf(alpha[(size_t)s * B + b] + final_logp[s] - mx);
    red[tid] = acc; __syncthreads();
    for (int w = 128; w > 0; w >>= 1) {
        if (tid < w) red[tid] += red[tid + w];
        __syncthreads();
    }
    if (tid == 0) per_seq[b] = mx + __logf(red[0]);
}

// ---------------------------------------------------------------------------
// K5: sum of the 64 per-sequence values via V_WMMA_F32_16X16X4_F32 with an
// all-ones B matrix (reduction-by-ones on the matrix pipe).  A is the 16x4
// f32 matrix holding the 64 values per the ISA layout: lanes 0-15 carry
// K=0,1 in VGPR0,1 for row M=lane; lanes 16-31 carry K=2,3 for M=lane-16.
// D[m][n] = sum_k A[m][k]; VGPR r of lane 0 holds M=r, of lane 16 holds M=8+r,
// so two lane-shuffles finish the 16-row combine.  Launched with 64 threads
// (2 full waves, EXEC all-ones as WMMA requires); wave 1 is redundant.
// ---------------------------------------------------------------------------
__global__ __launch_bounds__(64)
void sum_kernel(const float* __restrict__ per_seq, float* __restrict__ out) {
#ifdef HAVE_WMMA_F32X4
    const int lane = threadIdx.x & 31;
    const int Mrow = lane & 15;
    const int Klo  = (lane >> 4) * 2;            // 0 or 2
    v2f a;
    a.x = per_seq[Mrow * 4 + Klo + 0];
    a.y = per_seq[Mrow * 4 + Klo + 1];
    v2f ones;
    ones.x = 1.0f; ones.y = 1.0f;                // all-ones B: layout-invariant
    v8f c = {};
    c = __builtin_amdgcn_wmma_f32_16x16x4_f32(
        /*neg_a=*/false, a, /*neg_b=*/false, ones,
        /*c_mod=*/(short)0, c, /*reuse_a=*/false, /*reuse_b=*/false);
    float s = c[0] + c[1] + c[2] + c[3] + c[4] + c[5] + c[6] + c[7];
    float total = __shfl(s, 0, 32) + __shfl(s, 16, 32);  // M=0..7 + M=8..15
    if (threadIdx.x == 0) out[0] = total;
#else
    __shared__ float red[B];
    const int tid = threadIdx.x;
    red[tid] = per_seq[tid];
    __syncthreads();
    for (int w = B / 2; w > 0; w >>= 1) {
        if (tid < w) red[tid] += red[tid + w];
        __syncthreads();
    }
    if (tid == 0) out[0] = red[0];
#endif
}

// ---------------------------------------------------------------------------
extern "C" void kernel_launch(void* const* d_in, const int* in_sizes, int n_in,
                              void* d_out, int out_size, void* d_ws, size_t ws_size,
                              hipStream_t stream) {
    (void)in_sizes; (void)n_in; (void)out_size; (void)ws_size;

    const float* input      = (const float*)d_in[0];   // [T,B,D]
    const float* arc_logw   = (const float*)d_in[1];   // [A]
    const float* init_logp  = (const float*)d_in[2];   // [S]
    const float* final_logp = (const float*)d_in[3];   // [S]
    const int*   from_state = (const int*)d_in[4];     // [A]
    const int*   to_state   = (const int*)d_in[5];     // [A]
    const int*   pdf_id     = (const int*)d_in[6];     // [A]
    float* out = (float*)d_out;

    // Workspace layout (all 128B aligned): inputT | alpha | m | sum | per_seq
    char* ws = (char*)d_ws;
    const size_t inputT_bytes = (size_t)T * D * B * sizeof(float);   // 78.6 MB
    const size_t state_bytes  = (size_t)S * B * sizeof(float);       // 5.12 MB
    float*        inputT  = (float*)ws;
    float*        alpha   = (float*)(ws + inputT_bytes);
    unsigned int* m       = (unsigned int*)(ws + inputT_bytes + state_bytes);
    float*        sumbuf  = (float*)(ws + inputT_bytes + 2 * state_bytes);
    float*        per_seq = (float*)(ws + inputT_bytes + 3 * state_bytes);

    transpose_kernel<<<dim3(D / 32, B / 32, T), dim3(32, 8), 0, stream>>>(input, inputT);
    init_state_kernel<<<(S * B + 255) / 256, 256, 0, stream>>>(init_logp, alpha, m, sumbuf);

    const size_t ldsBytes = (size_t)D * HALF_B * sizeof(float);      // 256 KB (<=320 KB/WGP)
    const dim3 arcGrid(A / ARCS_PER_BLOCK, 2);
    const dim3 arcBlock(32, 8);
    const int finGrid = (S * B + 255) / 256;

    for (int t = 0; t < T; ++t) {
        const float* lt = inputT + (size_t)t * D * B;
        arc_kernel<0><<<arcGrid, arcBlock, ldsBytes, stream>>>(
            lt, alpha, arc_logw, from_state, to_state, pdf_id, m, sumbuf);
        arc_kernel<1><<<arcGrid, arcBlock, ldsBytes, stream>>>(
            lt, alpha, arc_logw, from_state, to_state, pdf_id, m, sumbuf);
        finalize_kernel<<<finGrid, 256, 0, stream>>>(m, sumbuf, alpha);
    }

    batch_lse_kernel<<<B, 256, 0, stream>>>(alpha, final_logp, per_seq);
    sum_kernel<<<1, B, 0, stream>>>(per_seq, out);
}